// SphericalBasisLayer_67035849556217
// MI455X (gfx1250) — compile-verified
//
#include <hip/hip_runtime.h>
#include <hip/hip_bf16.h>
#include <math.h>

#define NSPH 7
#define NRAD 6
#define NCOL 42   // NSPH*NRAD
#define TPB  256

typedef __attribute__((ext_vector_type(2))) float v2f;
typedef __attribute__((ext_vector_type(8))) float v8f;

// First 6 positive zeros of spherical Bessel j_l, l = 0..6
__constant__ double ZD[NSPH][NRAD] = {
  {3.141592653589793, 6.283185307179586, 9.424777960769380, 12.566370614359172, 15.707963267948966, 18.849555921538759},
  {4.493409457909064, 7.725251836937707, 10.904121659428899, 14.066193912831473, 17.220755271930768, 20.371302959287563},
  {5.763459196894550, 9.095011330476355, 12.322940970566582, 15.514603010886749, 18.689036355362822, 21.853874222709884},
  {6.987932000500519, 10.417118547379365, 13.698023153250246, 16.923621285214358, 20.121806174454734, 23.304246988939651},
  {8.182561452571243, 11.704907154571301, 15.039664707616036, 18.301255959541978, 21.525417733399770, 24.727565547835640},
  {9.355812111042747, 12.966530172775371, 16.354709639850598, 19.653152101821291, 22.904550647902713, 26.127750137810252},
  {10.512835408093997, 14.207392458842460, 17.647974906212970, 20.983463068644029, 24.262768347916406, 27.507868386156914}
};
__constant__ float ZF[NSPH][NRAD] = {
  {3.14159265f, 6.28318531f, 9.42477796f, 12.56637061f, 15.70796327f, 18.84955592f},
  {4.49340946f, 7.72525184f, 10.90412166f, 14.06619391f, 17.22075527f, 20.37130296f},
  {5.76345920f, 9.09501133f, 12.32294097f, 15.51460301f, 18.68903636f, 21.85387422f},
  {6.98793200f, 10.41711855f, 13.69802315f, 16.92362129f, 20.12180617f, 23.30424699f},
  {8.18256145f, 11.70490715f, 15.03966471f, 18.30125596f, 21.52541773f, 24.72756555f},
  {9.35581211f, 12.96653017f, 16.35470964f, 19.65315210f, 22.90455065f, 26.12775014f},
  {10.51283541f, 14.20739246f, 17.64797491f, 20.98346307f, 24.26276835f, 27.50786839f}
};
// sqrt((2l+1)/(4*pi))
__constant__ float SLC[NSPH] = {
  0.28209479f, 0.48860251f, 0.63078313f, 0.74635267f, 0.84628438f, 0.93560258f, 1.01710724f
};
// B fragments for V_WMMA_F32_16X16X4_F32 (4x16 tiles of monomial->Y_l coeff matrix).
// Layout: VGPR0 = rows (K, K+2) [lanes 0-15 | 16-31], VGPR1 = rows (K+1, K+3).
// BFRAG[0]: k=0 | k=2 ; BFRAG[1]: k=1 | k=3 ; BFRAG[2]: k=4 | k=6 ; BFRAG[3]: k=5 | k=7(=0)
__constant__ float BFRAG[4][32] = {
  { 0.28209479f, 0.0f, -0.31539157f, 0.0f, 0.31735664f, 0.0f, -0.31784601f, 0.0f,
    0,0,0,0,0,0,0,0,
    0.0f, 0.0f, 0.94617470f, 0.0f, -3.17356641f, 0.0f, 6.67476624f, 0.0f,
    0,0,0,0,0,0,0,0 },
  { 0.0f, 0.48860251f, 0.0f, -1.11952900f, 0.0f, 1.75425484f, 0.0f, 0.0f,
    0,0,0,0,0,0,0,0,
    0.0f, 0.0f, 0.0f, 1.86588166f, 0.0f, -8.18652257f, 0.0f, 0.0f,
    0,0,0,0,0,0,0,0 },
  { 0.0f, 0.0f, 0.0f, 0.0f, 3.70249414f, 0.0f, -20.02429872f, 0.0f,
    0,0,0,0,0,0,0,0,
    0.0f, 0.0f, 0.0f, 0.0f, 0.0f, 0.0f, 14.68448573f, 0.0f,
    0,0,0,0,0,0,0,0 },
  { 0.0f, 0.0f, 0.0f, 0.0f, 0.0f, 7.36787031f, 0.0f, 0.0f,
    0,0,0,0,0,0,0,0,
    0,0,0,0,0,0,0,0, 0,0,0,0,0,0,0,0 }
};

__device__ __forceinline__ float envelope_f(float x) {
  // p = ENV_EXPONENT+1 = 6 : 1/x - 28 x^5 + 48 x^6 - 21 x^7, zero for x >= 1
  float inv = 1.0f / x;
  float x2 = x * x, x4 = x2 * x2, x5 = x4 * x;
  float env = inv + x5 * (-28.0f + x * (48.0f - 21.0f * x));
  return (x < 1.0f) ? env : 0.0f;
}

__device__ __forceinline__ float sph_jl_f(float x, int l) {
  float s, c;
  s = sinf(x); c = cosf(x);
  float inv = 1.0f / x;
  float j0 = s * inv;
  if (l == 0) return j0;
  float j1 = (s * inv - c) * inv;
  float jm = j0, jl = j1;
  #pragma unroll
  for (int m = 1; m < 7; ++m) {
    if (m >= l) break;
    float t = (float)(2 * m + 1) * inv * jl - jm;
    jm = jl; jl = t;
  }
  return jl;
}

// 42 threads: NORM[l,n] = sqrt(2)/|j_{l+1}(z_{l,n})| in float64 (matches reference precompute)
__global__ void norms_kernel(float* __restrict__ norm) {
  int t = threadIdx.x;
  if (t >= NCOL) return;
  int l = t / NRAD, n = t % NRAD;
  double z = ZD[l][n];
  double s = sin(z), c = cos(z), inv = 1.0 / z;
  double jm = s * inv;
  double jl = (s * inv - c) * inv;
  for (int m = 1; m <= l; ++m) {           // after loop: jl = j_{l+1}(z)
    double tt = (double)(2 * m + 1) * inv * jl - jm;
    jm = jl; jl = tt;
  }
  norm[t] = (float)(1.4142135623730951 / fabs(jl));
}

// Per-edge radial table: Eenv[e, l*6+n] = env(x)*NORM[l,n]*j_l(x*Z[l,n]); 84MB, L2-resident.
__global__ void edge_rbf_kernel(const float* __restrict__ d, const float* __restrict__ norm,
                                float* __restrict__ Eenv, int n_edges) {
  int e = blockIdx.x * blockDim.x + threadIdx.x;
  if (e >= n_edges) return;
  float x = __builtin_nontemporal_load(d + e) * 0.2f;
  float env = envelope_f(x);
  v2f* out2 = (v2f*)(Eenv + (size_t)e * NCOL);
  #pragma unroll
  for (int l = 0; l < NSPH; ++l) {
    float r[NRAD];
    #pragma unroll
    for (int n = 0; n < NRAD; ++n)
      r[n] = env * norm[l * NRAD + n] * sph_jl_f(x * ZF[l][n], l);
    #pragma unroll
    for (int p = 0; p < NRAD / 2; ++p) {
      v2f v; v[0] = r[2 * p]; v[1] = r[2 * p + 1];
      out2[l * 3 + p] = v;
    }
  }
}

// Per-triplet: Y_l via WMMA (monomial basis x coeff matrix), then gather*multiply, NT stores.
__global__ void triplet_kernel(const float* __restrict__ Angles, const int* __restrict__ idx,
                               const float* __restrict__ Eenv, float* __restrict__ out,
                               int n_trip) {
  __shared__ float Ylds[TPB * 8];
  __shared__ int   elds[TPB];
  const int tid  = threadIdx.x;
  const int lane = tid & 31;
  const int wid  = tid >> 5;
  const int tblk = blockIdx.x * TPB;

  // ---- phase 1: cos, edge gather index, Y_l via 2x chained V_WMMA_F32_16X16X4_F32 ----
  int t  = tblk + wid * 32 + lane;
  int tc = (t < n_trip) ? t : (n_trip - 1);
  float ct = cosf(__builtin_nontemporal_load(Angles + tc));
  int e = __builtin_nontemporal_load(idx + tc);
  elds[wid * 32 + lane] = e;
  // warm the gathered row (2 cachelines) -> global_prefetch_b8
  __builtin_prefetch(Eenv + (size_t)e * NCOL, 0, 3);
  __builtin_prefetch(Eenv + (size_t)e * NCOL + 32, 0, 3);

  v2f bA; bA[0] = BFRAG[0][lane]; bA[1] = BFRAG[1][lane];
  v2f bB; bB[0] = BFRAG[2][lane]; bB[1] = BFRAG[3][lane];
  const bool hi  = lane >= 16;
  const int  col = hi ? (lane - 16) : lane;

  #pragma unroll
  for (int tile = 0; tile < 2; ++tile) {
    // rows M=0..15 of this tile come from lanes (tile*16 + 0..15)
    float cr = __shfl(ct, (tile << 4) | (lane & 15), 32);
    float c2 = cr * cr, c3 = c2 * cr, c4 = c2 * c2, c5 = c4 * cr, c6 = c4 * c2;
    // A 16x4 layout: VGPR0 = K0|K2, VGPR1 = K1|K3 (second wmma: K4|K6, K5|K7)
    v2f aA; aA[0] = hi ? c2 : 1.0f; aA[1] = hi ? c3 : cr;
    v2f aB; aB[0] = hi ? c6 : c4;   aB[1] = hi ? 0.0f : c5;
    v8f acc = {};
    acc = __builtin_amdgcn_wmma_f32_16x16x4_f32(false, aA, false, bA, (short)0, acc, false, false);
    acc = __builtin_amdgcn_wmma_f32_16x16x4_f32(false, aB, false, bB, (short)0, acc, false, false);
    // D layout: VGPR j -> (M=j, N=lane) lanes 0-15 ; (M=j+8, N=lane-16) lanes 16-31
    if (col < NSPH) {
      int mbase = wid * 32 + (tile << 4) + (hi ? 8 : 0);
      #pragma unroll
      for (int j = 0; j < 8; ++j)
        Ylds[(mbase + j) * 8 + col] = acc[j];
    }
  }
  __syncthreads();

  // ---- phase 2: coalesced float2 gather-multiply-store (21 f2 per triplet) ----
  int ntb   = n_trip - tblk; if (ntb > TPB) ntb = TPB;
  int count = ntb * (NCOL / 2);
  const v2f* E2 = (const v2f*)Eenv;
  v2f* O2 = (v2f*)out + (size_t)tblk * (NCOL / 2);
  for (int i = tid; i < count; i += TPB) {
    int tl  = i / 21;
    int c2i = i - tl * 21;          // float2 column; covers cols (2*c2i, 2*c2i+1), same l
    int l   = c2i / 3;
    int ee  = elds[tl];
    v2f ev  = E2[(size_t)ee * 21 + c2i];   // L2-resident gather
    float y = Ylds[tl * 8 + l];
    v2f o; o[0] = ev[0] * y; o[1] = ev[1] * y;
    __builtin_nontemporal_store(o, O2 + i);  // don't thrash L2 with the 672MB stream
  }
}

// Fallback if workspace is too small: fully fused per-triplet (no scratch needed).
__global__ void fused_fallback_kernel(const float* __restrict__ d, const float* __restrict__ Angles,
                                      const int* __restrict__ idx, float* __restrict__ out,
                                      int n_trip) {
  int t = blockIdx.x * blockDim.x + threadIdx.x;
  if (t >= n_trip) return;
  int e = idx[t];
  float x   = d[e] * 0.2f;
  float env = envelope_f(x);
  float ct  = cosf(Angles[t]);
  float* o  = out + (size_t)t * NCOL;
  float Pp = 1.0f, Pc = 1.0f;
  #pragma unroll
  for (int l = 0; l < NSPH; ++l) {
    if (l == 1) { Pp = Pc; Pc = ct; }
    else if (l >= 2) {
      float Pn = ((float)(2 * l - 1) * ct * Pc - (float)(l - 1) * Pp) / (float)l;
      Pp = Pc; Pc = Pn;
    }
    float Y = SLC[l] * Pc;
    #pragma unroll
    for (int n = 0; n < NRAD; ++n) {
      float nrm = 1.41421356f / fabsf(sph_jl_f(ZF[l][n], l + 1));
      o[l * NRAD + n] = env * nrm * sph_jl_f(x * ZF[l][n], l) * Y;
    }
  }
}

extern "C" void kernel_launch(void* const* d_in, const int* in_sizes, int n_in,
                              void* d_out, int out_size, void* d_ws, size_t ws_size,
                              hipStream_t stream) {
  const float* d      = (const float*)d_in[0];
  const float* Angles = (const float*)d_in[1];
  const int*   idx    = (const int*)d_in[2];
  float* out = (float*)d_out;
  int n_edges = in_sizes[0];
  int n_trip  = in_sizes[1];

  size_t need = 256 + (size_t)n_edges * NCOL * sizeof(float);
  if (ws_size >= need) {
    float* normt = (float*)d_ws;
    float* Eenv  = (float*)((char*)d_ws + 256);
    norms_kernel<<<1, 64, 0, stream>>>(normt);
    edge_rbf_kernel<<<(n_edges + TPB - 1) / TPB, TPB, 0, stream>>>(d, normt, Eenv, n_edges);
    triplet_kernel<<<(n_trip + TPB - 1) / TPB, TPB, 0, stream>>>(Angles, idx, Eenv, out, n_trip);
  } else {
    fused_fallback_kernel<<<(n_trip + TPB - 1) / TPB, TPB, 0, stream>>>(d, Angles, idx, out, n_trip);
  }
}